// LIFMaxPool2d_46626164965950
// MI455X (gfx1250) — compile-verified
//
#include <hip/hip_runtime.h>

// LIF + 2x2 max-pool, fully fused, single pass, HBM-streaming kernel.
// B=16 C=64 H=256 W=256 -> out 16x64x128x128.
//
// Precomputed LIF constants (norse defaults):
//   i' = syn*(1 - dt*tau_syn_inv) + x  = syn*0.8 + x
//   v' = v + dt*tau_mem_inv*((0 - v) + i') = v*0.9 + 0.1*i'
//   spike = (v' >= 1.0)
// Pool(2x2) of binary spikes == (max of the 4 v' values >= 1.0).

typedef float v4f __attribute__((ext_vector_type(4)));
typedef float v2f __attribute__((ext_vector_type(2)));

#define LIF_W   256
#define LIF_H   256
#define LIF_OW  128
#define LIF_OH  128
// per-thread tile: 4 input cols x 2 input rows -> 2 pooled outputs
#define QUADS_PER_ROW (LIF_W / 4)   // 64

__global__ __launch_bounds__(256) void lif_maxpool_kernel(
    const float* __restrict__ x,     // input_signal
    const float* __restrict__ mem,   // membrane
    const float* __restrict__ syn,   // synaptic
    float* __restrict__ out,
    int total_threads)
{
    const int tid = blockIdx.x * blockDim.x + threadIdx.x;
    if (tid >= total_threads) return;

    // tid -> (bc, oh, quad). quad indexes groups of 4 input columns.
    const int quad = tid & (QUADS_PER_ROW - 1);          // [0,64)
    const int oh   = (tid >> 6) & (LIF_OH - 1);          // [0,128)
    const int bc   = tid >> 13;                          // 64*128 = 2^13

    const long base = (long)bc * (LIF_H * LIF_W)
                    + (long)(2 * oh) * LIF_W
                    + (long)(quad * 4);

    // Non-temporal b128 loads: streamed once, never reused -> bypass cache retention.
    const v4f x0 = __builtin_nontemporal_load((const v4f*)(x   + base));
    const v4f m0 = __builtin_nontemporal_load((const v4f*)(mem + base));
    const v4f s0 = __builtin_nontemporal_load((const v4f*)(syn + base));
    const v4f x1 = __builtin_nontemporal_load((const v4f*)(x   + base + LIF_W));
    const v4f m1 = __builtin_nontemporal_load((const v4f*)(mem + base + LIF_W));
    const v4f s1 = __builtin_nontemporal_load((const v4f*)(syn + base + LIF_W));

    const float kSyn = 1.0f - 0.001f * 200.0f;   // 0.8
    const float kMem = 1.0f - 0.001f * 100.0f;   // 0.9
    const float kDtM = 0.001f * 100.0f;          // 0.1

    // v' for all 8 elements: v = fma(mem, 0.9, 0.1 * fma(syn, 0.8, x))
    v4f v0, v1;
    #pragma unroll
    for (int k = 0; k < 4; ++k) {
        const float i0 = __builtin_fmaf(s0[k], kSyn, x0[k]);
        const float i1 = __builtin_fmaf(s1[k], kSyn, x1[k]);
        v0[k] = __builtin_fmaf(m0[k], kMem, kDtM * i0);
        v1[k] = __builtin_fmaf(m1[k], kMem, kDtM * i1);
    }

    // Pool windows: cols {0,1} and cols {2,3}, across both rows.
    const float p0 = fmaxf(fmaxf(v0[0], v0[1]), fmaxf(v1[0], v1[1]));
    const float p1 = fmaxf(fmaxf(v0[2], v0[3]), fmaxf(v1[2], v1[3]));

    v2f r;
    r[0] = (p0 >= 1.0f) ? 1.0f : 0.0f;
    r[1] = (p1 >= 1.0f) ? 1.0f : 0.0f;

    const long obase = (long)bc * (LIF_OH * LIF_OW)
                     + (long)oh * LIF_OW
                     + (long)(quad * 2);
    __builtin_nontemporal_store(r, (v2f*)(out + obase));
}

extern "C" void kernel_launch(void* const* d_in, const int* in_sizes, int n_in,
                              void* d_out, int out_size, void* d_ws, size_t ws_size,
                              hipStream_t stream) {
    (void)in_sizes; (void)n_in; (void)d_ws; (void)ws_size;
    const float* x   = (const float*)d_in[0];   // input_signal
    const float* mem = (const float*)d_in[1];   // membrane
    const float* syn = (const float*)d_in[2];   // synaptic
    float* out = (float*)d_out;

    // Each thread produces 2 pooled outputs.
    const int total_threads = out_size / 2;     // 16*64*128*128 / 2 = 8,388,608
    const int block = 256;                      // 8 wave32 per block
    const int grid  = (total_threads + block - 1) / block;

    lif_maxpool_kernel<<<grid, block, 0, stream>>>(x, mem, syn, out, total_threads);
}